// PatchEmbedding_35725537968368
// MI455X (gfx1250) — compile-verified
//
#include <hip/hip_runtime.h>
#include <hip/hip_bf16.h>

// PatchEmbedding via vocab factorization:
//   Phase 1 (WMMA): M2[p, v, dout] = sum_d W[dout, p*768+d] * emb[v, d]
//       GEMM [172800 x 768] x [768 x 16(pad from 11)]  -- streams W (531 MB) once
//   Phase 2 (gather): out[b, n+1, dout] = bias[dout] + pos[n+1,dout]
//                                       + sum_p M2[p, tok[b,n,p], dout]
//       M2 (7.6 MB) stays L2-resident.

typedef __attribute__((ext_vector_type(2))) float v2f;
typedef __attribute__((ext_vector_type(8))) float v8f;

#define D_EMBED 768
#define PP      225          // P*P
#define VOCAB   11
#define MROWS   (768 * 225)  // 172800 rows of W viewed as [rows x 768]
#define NTILES  (MROWS / 16) // 10800 -> 1350 blocks x 8 waves

__global__ __launch_bounds__(256)
void pe_phase1_wmma(const float* __restrict__ W,
                    const float* __restrict__ emb,
                    float* __restrict__ M2)
{
    // B operand staged in LDS, pair-major so each fragment is one b64 read:
    // entry (k, v) stored at ldsB[(k>>1)*32 + v*2 + (k&1)]   (384*32 floats = 48 KB)
    __shared__ float ldsB[(D_EMBED / 2) * 32];

    const int t = threadIdx.x;
    for (int idx = t; idx < D_EMBED * 16; idx += 256) {
        int k = idx >> 4;
        int v = idx & 15;
        float val = (v < VOCAB) ? emb[v * D_EMBED + k] : 0.0f;
        ldsB[(k >> 1) * 32 + v * 2 + (k & 1)] = val;
    }
    __syncthreads();

    const int wave = t >> 5;
    const int lane = t & 31;
    const int tile = blockIdx.x * 8 + wave;   // grid = 1350 -> tile < 10800, no tail
    const int m0   = tile * 16;
    const int half = lane >> 4;               // 0: K=0,1   1: K=2,3
    const int l16  = lane & 15;

    // A: lane holds row (m0 + l16), columns (k0 + 2*half, k0 + 2*half + 1)
    const float* Abase = W + (size_t)(m0 + l16) * D_EMBED + half * 2;
    // B: lane holds emb[v=l16][k0 + 2*half .. +1]
    const float* Bbase = ldsB + half * 32 + l16 * 2;

    v8f acc = {};
    #pragma unroll 4
    for (int k0 = 0; k0 < D_EMBED; k0 += 4) {
        v2f a = *(const v2f*)(Abase + k0);
        v2f b = *(const v2f*)(Bbase + (k0 >> 1) * 32);
        acc = __builtin_amdgcn_wmma_f32_16x16x4_f32(
                  /*neg_a=*/false, a, /*neg_b=*/false, b,
                  /*c_mod=*/(short)0, acc, /*reuse_a=*/false, /*reuse_b=*/false);
    }

    // C/D layout: VGPR j -> M = j (lanes 0-15) or M = 8+j (lanes 16-31); N = lane&15
    const int v = l16;
    if (v < VOCAB) {
        #pragma unroll
        for (int j = 0; j < 8; ++j) {
            int r    = m0 + j + half * 8;     // row of W-view = dout*225 + p
            int dout = r / PP;
            int p    = r - dout * PP;
            M2[(size_t)(p * VOCAB + v) * D_EMBED + dout] = acc[j];
        }
    }
}

__global__ __launch_bounds__(256)
void pe_phase2_gather(const int* __restrict__ x,
                      const float* __restrict__ M2,
                      const float* __restrict__ bias,
                      const float* __restrict__ cls,
                      const float* __restrict__ pos,
                      float* __restrict__ out)
{
    __shared__ int tok[PP];

    const int blk = blockIdx.x;        // 0 .. 64*24-1
    const int b   = blk / 24;
    const int n   = blk % 24;
    const int pi  = n / 12;            // GW/P = 12 patches per row
    const int pj  = n % 12;
    const int t   = threadIdx.x;

    if (t < PP) {
        int r = t / 15;
        int c = t - r * 15;
        tok[t] = x[b * (30 * 180) + (pi * 15 + r) * 180 + (pj * 15 + c)];
    }
    __syncthreads();

    float acc0 = bias[t];
    float acc1 = bias[t + 256];
    float acc2 = bias[t + 512];
    #pragma unroll 5
    for (int p = 0; p < PP; ++p) {
        const float* row = M2 + (size_t)(p * VOCAB + tok[p]) * D_EMBED;
        acc0 += row[t];
        acc1 += row[t + 256];
        acc2 += row[t + 512];
    }

    float*       orow = out + ((size_t)b * 25 + (n + 1)) * D_EMBED;
    const float* prow = pos + (n + 1) * D_EMBED;
    orow[t]       = acc0 + prow[t];
    orow[t + 256] = acc1 + prow[t + 256];
    orow[t + 512] = acc2 + prow[t + 512];

    if (n == 0) {  // cls row: out[b, 0, :] = cls + pos[0, :]
        float* o0 = out + (size_t)b * 25 * D_EMBED;
        o0[t]       = cls[t]       + pos[t];
        o0[t + 256] = cls[t + 256] + pos[t + 256];
        o0[t + 512] = cls[t + 512] + pos[t + 512];
    }
}

extern "C" void kernel_launch(void* const* d_in, const int* in_sizes, int n_in,
                              void* d_out, int out_size, void* d_ws, size_t ws_size,
                              hipStream_t stream) {
    const int*   x    = (const int*)d_in[0];    // [64, 30, 180]
    const float* emb  = (const float*)d_in[1];  // [11, 768]
    const float* W    = (const float*)d_in[2];  // [768, 172800]
    const float* bias = (const float*)d_in[3];  // [768]
    const float* cls  = (const float*)d_in[4];  // [1, 1, 768]
    const float* pos  = (const float*)d_in[5];  // [1, 25, 768]
    float*       out  = (float*)d_out;          // [64, 25, 768]
    float*       M2   = (float*)d_ws;           // [225, 11, 768] = 7.6 MB

    pe_phase1_wmma<<<NTILES / 8, 256, 0, stream>>>(W, emb, M2);
    pe_phase2_gather<<<64 * 24, 256, 0, stream>>>(x, M2, bias, cls, pos, out);
}